// SwitchGate_86517821214173
// MI455X (gfx1250) — compile-verified
//
#include <hip/hip_runtime.h>
#include <math.h>

// SwitchGate router for MI455X (gfx1250, wave32).
// logits = inp(16384x4096) @ W^T(4096x64) + b ; softmax ; top1 ; aux loss.
// capacity (39322) > T (16384) => pruning is a no-op.
//
// Block = 256 threads = 8 waves, owns 32 tokens (two 16-row WMMA groups).
// Waves split K=4096 into 512-wide chunks; each wave computes 2x4 fp32 WMMA
// accumulator tiles (32 tokens x 64 experts) and reduces via ds_add_f32.
// B tiles (W) are reused across both token groups -> halves L2 W-traffic
// vs one group per block (512 MB total vs 1 GB) and gives 6 loads : 8 WMMAs
// per 4-K step.

typedef float v2f __attribute__((ext_vector_type(2)));
typedef float v8f __attribute__((ext_vector_type(8)));

#define T_TOK 16384
#define D_DIM 4096
#define E_EXP 64
#define KCHUNK 512          // K per wave: 8 waves * 512 = 4096
#define TOK_BLK 32          // tokens per block (two 16-row groups)

__global__ __launch_bounds__(256) void gate_main_kernel(
    const float* __restrict__ inp,   // [T, D]
    const float* __restrict__ W,     // [E, D]
    const float* __restrict__ bias,  // [E]
    float* __restrict__ out,         // [T idx | T score | 1 loss]
    float* __restrict__ ws)          // [64 scoresum | 64 count]
{
    __shared__ float lds_logits[TOK_BLK * E_EXP];  // 8 KB
    __shared__ float lds_esum[E_EXP];
    __shared__ float lds_ecnt[E_EXP];
    __shared__ float lds_b[E_EXP];

    const int tid  = threadIdx.x;
    const int wave = tid >> 5;
    const int lane = tid & 31;
    const int half = lane >> 4;   // 0: lanes 0-15, 1: lanes 16-31
    const int lr   = lane & 15;
    const int m0   = blockIdx.x * TOK_BLK;   // token base for this block

    for (int i = tid; i < TOK_BLK * E_EXP; i += 256) lds_logits[i] = 0.0f;
    if (tid < E_EXP) {
        lds_esum[tid] = 0.0f;
        lds_ecnt[tid] = 0.0f;
        lds_b[tid]    = bias[tid];
    }
    __syncthreads();

    // ---- GEMM phase: this wave covers K in [k_base, k_base + KCHUNK) ----
    // A 16x4 f32 tile: lanes 0-15 hold row M=lane, VGPR pair = K,K+1;
    // lanes 16-31 hold K+2,K+3  -> one contiguous float2 load per step.
    // B 4x16 f32 tile (K x N): lane = expert column, same K split.
    const int k_base = wave * KCHUNK;
    const float* Aptr0 = inp + (size_t)(m0 + lr)      * D_DIM + k_base + 2 * half;
    const float* Aptr1 = inp + (size_t)(m0 + 16 + lr) * D_DIM + k_base + 2 * half;
    const float* Bptr  = W   + (size_t)lr             * D_DIM + k_base + 2 * half;

    v8f acc00 = {}, acc01 = {}, acc02 = {}, acc03 = {};   // group 0 (tokens m0..m0+15)
    v8f acc10 = {}, acc11 = {}, acc12 = {}, acc13 = {};   // group 1 (tokens m0+16..m0+31)

    #pragma unroll 4
    for (int k = 0; k < KCHUNK; k += 4) {
        v2f a0 = *(const v2f*)(Aptr0 + k);
        v2f a1 = *(const v2f*)(Aptr1 + k);
        v2f b0 = *(const v2f*)(Bptr + k);
        v2f b1 = *(const v2f*)(Bptr + (size_t)16 * D_DIM + k);
        v2f b2 = *(const v2f*)(Bptr + (size_t)32 * D_DIM + k);
        v2f b3 = *(const v2f*)(Bptr + (size_t)48 * D_DIM + k);
        // D = A x B + C on the fp32 matrix pipe (V_WMMA_F32_16X16X4_F32)
        acc00 = __builtin_amdgcn_wmma_f32_16x16x4_f32(false, a0, false, b0,
                                                      (short)0, acc00, false, false);
        acc01 = __builtin_amdgcn_wmma_f32_16x16x4_f32(false, a0, false, b1,
                                                      (short)0, acc01, false, false);
        acc02 = __builtin_amdgcn_wmma_f32_16x16x4_f32(false, a0, false, b2,
                                                      (short)0, acc02, false, false);
        acc03 = __builtin_amdgcn_wmma_f32_16x16x4_f32(false, a0, false, b3,
                                                      (short)0, acc03, false, false);
        acc10 = __builtin_amdgcn_wmma_f32_16x16x4_f32(false, a1, false, b0,
                                                      (short)0, acc10, false, false);
        acc11 = __builtin_amdgcn_wmma_f32_16x16x4_f32(false, a1, false, b1,
                                                      (short)0, acc11, false, false);
        acc12 = __builtin_amdgcn_wmma_f32_16x16x4_f32(false, a1, false, b2,
                                                      (short)0, acc12, false, false);
        acc13 = __builtin_amdgcn_wmma_f32_16x16x4_f32(false, a1, false, b3,
                                                      (short)0, acc13, false, false);
    }

    // ---- Cross-wave K reduction into LDS logits[token][expert] ----
    // C/D layout: VGPR r holds M=r (lanes 0-15) and M=8+r (lanes 16-31),
    // N = lane % 16.
    #pragma unroll
    for (int r = 0; r < 8; ++r) {
        const int tok0 = 8 * half + r;
        const int tok1 = 16 + tok0;
        atomicAdd(&lds_logits[tok0 * E_EXP +  0 + lr], acc00[r]);
        atomicAdd(&lds_logits[tok0 * E_EXP + 16 + lr], acc01[r]);
        atomicAdd(&lds_logits[tok0 * E_EXP + 32 + lr], acc02[r]);
        atomicAdd(&lds_logits[tok0 * E_EXP + 48 + lr], acc03[r]);
        atomicAdd(&lds_logits[tok1 * E_EXP +  0 + lr], acc10[r]);
        atomicAdd(&lds_logits[tok1 * E_EXP + 16 + lr], acc11[r]);
        atomicAdd(&lds_logits[tok1 * E_EXP + 32 + lr], acc12[r]);
        atomicAdd(&lds_logits[tok1 * E_EXP + 48 + lr], acc13[r]);
    }
    __syncthreads();

    // ---- Softmax / top-1 / expert stats: one lane per token ----
    if (tid < TOK_BLK) {
        const int tok = tid;
        float mx = -INFINITY;
        int   bi = 0;
        #pragma unroll 8
        for (int e = 0; e < E_EXP; ++e) {
            float v = lds_logits[tok * E_EXP + e] + lds_b[e];
            if (v > mx) { mx = v; bi = e; }   // strict > keeps lowest index on ties
        }
        float sum = 0.0f;
        #pragma unroll 8
        for (int e = 0; e < E_EXP; ++e) {
            float v = lds_logits[tok * E_EXP + e] + lds_b[e];
            sum += expf(v - mx);
        }
        const float inv = 1.0f / sum;
        // outputs (idx as float: values in [-1, 63] are exact)
        out[m0 + tok]          = (float)bi;
        out[T_TOK + m0 + tok]  = inv;         // top1 prob = exp(mx)/sum
        // expert score sums (prob_expert numerator)
        #pragma unroll 8
        for (int e = 0; e < E_EXP; ++e) {
            float v = lds_logits[tok * E_EXP + e] + lds_b[e];
            atomicAdd(&lds_esum[e], expf(v - mx) * inv);
        }
        atomicAdd(&lds_ecnt[bi], 1.0f);
    }
    __syncthreads();

    // ---- Flush per-block stats to global workspace ----
    if (tid < E_EXP) {
        atomicAdd(&ws[tid],         lds_esum[tid]);
        atomicAdd(&ws[E_EXP + tid], lds_ecnt[tid]);
    }
}

__global__ __launch_bounds__(128) void gate_init_kernel(float* __restrict__ ws)
{
    if (threadIdx.x < 2 * E_EXP) ws[threadIdx.x] = 0.0f;
}

__global__ __launch_bounds__(64) void gate_loss_kernel(
    const float* __restrict__ ws, float* __restrict__ out)
{
    __shared__ float red[E_EXP];
    const int e = threadIdx.x;
    red[e] = ws[e] * ws[E_EXP + e];   // scoresum[e] * count[e]
    __syncthreads();
    #pragma unroll
    for (int s = 32; s > 0; s >>= 1) {
        if (e < s) red[e] += red[e + s];
        __syncthreads();
    }
    if (e == 0) {
        const float scale = (float)E_EXP / ((float)T_TOK * (float)T_TOK);
        out[2 * T_TOK] = red[0] * scale;
    }
}

extern "C" void kernel_launch(void* const* d_in, const int* in_sizes, int n_in,
                              void* d_out, int out_size, void* d_ws, size_t ws_size,
                              hipStream_t stream)
{
    const float* inp  = (const float*)d_in[0];   // [T, D] fp32
    const float* W    = (const float*)d_in[1];   // [E, D] fp32
    const float* bias = (const float*)d_in[2];   // [E]    fp32
    float* out = (float*)d_out;                  // [T | T | 1] fp32
    float* ws  = (float*)d_ws;                   // >= 128 floats

    gate_init_kernel<<<1, 128, 0, stream>>>(ws);
    gate_main_kernel<<<T_TOK / TOK_BLK, 256, 0, stream>>>(inp, W, bias, out, ws);
    gate_loss_kernel<<<1, 64, 0, stream>>>(ws, out);
}